// MpnnDuvenaud_9655086481677
// MI455X (gfx1250) — compile-verified
//
#include <hip/hip_runtime.h>
#include <hip/hip_bf16.h>
#include <math.h>

#define Bq   256
#define Nn   64
#define DH   256
#define DE   64
#define OUT  256
#define ND   11
#define RO   512
#define TGT  12
#define DIN  320               // DH + DE
#define NODES (Bq * Nn)        // 16384

typedef __attribute__((ext_vector_type(2))) float v2f;
typedef __attribute__((ext_vector_type(8))) float v8f;

__device__ __forceinline__ v8f wmma_f32x4(v2f a, v2f b, v8f c) {
  // D = A(16x4 f32) * B(4x16 f32) + C(16x16 f32)
  return __builtin_amdgcn_wmma_f32_16x16x4_f32(false, a, false, b, (short)0, c,
                                               false, false);
}

// Async LDS copy (CDNA5): moves 16B global -> LDS, tracked by ASYNCcnt.
// IOFFSET applies to BOTH the LDS and global address, so one row (64B) is
// staged with four instructions sharing the same base registers.
__device__ __forceinline__ void async_row64B(uint32_t lds, const float* g) {
  asm volatile("global_load_async_to_lds_b128 %0, %1, off"
               :: "v"(lds), "v"(g) : "memory");
  asm volatile("global_load_async_to_lds_b128 %0, %1, off offset:16"
               :: "v"(lds), "v"(g) : "memory");
  asm volatile("global_load_async_to_lds_b128 %0, %1, off offset:32"
               :: "v"(lds), "v"(g) : "memory");
  asm volatile("global_load_async_to_lds_b128 %0, %1, off offset:48"
               :: "v"(lds), "v"(g) : "memory");
}

__device__ __forceinline__ void wait_async0() {
  asm volatile("s_wait_asynccnt 0" ::: "memory");
}

// ---------------- init: zero h1,h2 (contiguous), acc, degree counters -------
__global__ void k_init(float* h12, float* acc, int* cnt) {
  size_t i = (size_t)blockIdx.x * blockDim.x + threadIdx.x;
  size_t stride = (size_t)gridDim.x * blockDim.x;
  const size_t nh = (size_t)2 * NODES * OUT;
  for (size_t j = i; j < nh; j += stride) h12[j] = 0.f;
  const size_t na = (size_t)Bq * RO;
  for (size_t j = i; j < na; j += stride) acc[j] = 0.f;
  if (i < ND) cnt[i] = 0;
}

// ---------------- degree computation + compaction by degree -----------------
__global__ void k_deg(const float* __restrict__ g, int* cnt, int* perm) {
  int node = blockIdx.x * blockDim.x + threadIdx.x;
  if (node >= NODES) return;
  const float* row = g + (size_t)node * Nn;
  int deg = 0;
  for (int w = 0; w < Nn; ++w) deg += (row[w] != 0.f) ? 1 : 0;
  if (deg < ND) {
    int slot = atomicAdd(&cnt[deg], 1);   // order-free: result per node invariant
    perm[deg * NODES + slot] = node;
  }
}

// ---------------- m_e[v,:] = sum_w g[v,w] * e[v,w,:]  (g is 0/1, ~8% dense) -
__global__ __launch_bounds__(64) void k_me(const float* __restrict__ g,
                                           const float* __restrict__ e,
                                           float* __restrict__ mbuf) {
  int node = blockIdx.x;
  int t = threadIdx.x;
  const float* grow = g + (size_t)node * Nn;
  const float* erow = e + (size_t)node * Nn * DE;
  float acc = 0.f;
  for (int w = 0; w < Nn; ++w) {
    float gv = grow[w];                 // uniform across block
    if (gv != 0.f) acc += gv * erow[(size_t)w * DE + t];  // adding 0 skipped: exact
  }
  mbuf[(size_t)node * DIN + DH + t] = acc;
}

// ---------------- m_h = g @ h  (per batch 64x64 * 64x256), f32 WMMA ---------
__global__ __launch_bounds__(32) void k_mh(const float* __restrict__ g,
                                           const float* __restrict__ hprev,
                                           float* __restrict__ mbuf) {
  int ct = blockIdx.x;           // 16 col tiles of 16
  int rt = blockIdx.y;           // 4 row tiles of 16
  int b  = blockIdx.z;           // 256 batches
  int lane = threadIdx.x;
  int n = lane & 15, half = lane >> 4;
  const float* A  = g     + (size_t)b * Nn * Nn;   // 64x64
  const float* Bm = hprev + (size_t)b * Nn * DH;   // 64x256
  int row0 = rt * 16, col0 = ct * 16;
  v8f c = {};
  for (int k0 = 0; k0 < Nn; k0 += 4) {
    int k = k0 + 2 * half;
    v2f a, bb;
    a[0]  = A[(size_t)(row0 + n) * Nn + k];
    a[1]  = A[(size_t)(row0 + n) * Nn + k + 1];
    bb[0] = Bm[(size_t)(k)     * DH + col0 + n];
    bb[1] = Bm[(size_t)(k + 1) * DH + col0 + n];
    c = wmma_f32x4(a, bb, c);
  }
  for (int r = 0; r < 8; ++r) {
    int row = row0 + r + 8 * half;
    mbuf[(size_t)(b * Nn + row) * DIN + col0 + n] = c[r];
  }
}

// ------- degree-grouped update: h = sigmoid(m[perm] @ W[d]), f32 WMMA -------
// One wave stages the 320x16 W[d] column slab into LDS (async copy engine),
// then sweeps the degree group's row tiles (strided by gridDim.y) against it.
__global__ __launch_bounds__(32) void k_update(const float* __restrict__ mbuf,
                                               const float* __restrict__ Wu,
                                               const int* __restrict__ cnt,
                                               const int* __restrict__ perm,
                                               float* __restrict__ hnext) {
  __shared__ float sW[DIN * 16];       // 20 KB: W[d][k][col0..col0+15]
  int ct = blockIdx.x;                 // 16 col tiles
  int sg = blockIdx.y;                 // 8 strided row-tile groups
  int d  = blockIdx.z;                 // 11 degrees
  int cd = cnt[d];
  int ntiles = (cd + 15) >> 4;
  if (sg >= ntiles) return;            // uniform early-exit
  int lane = threadIdx.x;
  int n = lane & 15, half = lane >> 4;
  int col0 = ct * 16;

  const float* W = Wu + (size_t)d * DIN * OUT + col0;
  for (int k = lane; k < DIN; k += 32)                 // 10 rows per lane
    async_row64B((uint32_t)(uintptr_t)&sW[k * 16], W + (size_t)k * OUT);
  wait_async0();                                       // single wave: no barrier

  for (int rt = sg; rt < ntiles; rt += gridDim.y) {
    int row0 = rt * 16;
    int ra = row0 + n;
    int nodeA = (ra < cd) ? perm[d * NODES + ra] : -1;
    v8f c = {};
    for (int k0 = 0; k0 < DIN; k0 += 4) {
      int k = k0 + 2 * half;
      v2f a, bb;
      a[0]  = (nodeA >= 0) ? mbuf[(size_t)nodeA * DIN + k]     : 0.f;
      a[1]  = (nodeA >= 0) ? mbuf[(size_t)nodeA * DIN + k + 1] : 0.f;
      bb[0] = sW[(k)     * 16 + n];
      bb[1] = sW[(k + 1) * 16 + n];
      c = wmma_f32x4(a, bb, c);
    }
    for (int r = 0; r < 8; ++r) {
      int row = row0 + r + 8 * half;
      if (row < cd) {
        int node = perm[d * NODES + row];
        float x = c[r];
        hnext[(size_t)node * OUT + col0 + n] = 1.f / (1.f + expf(-x));
      }
    }
  }
}

// ---------------- readout z = hl @ R[l]  (16384x256 * 256x512), f32 WMMA ----
// One wave stages the 256x16 R column slab (16 KB LDS, async copy engine) and
// reuses it across 16 row tiles (256 rows).
__global__ __launch_bounds__(32) void k_readout(const float* __restrict__ hl,
                                                const float* __restrict__ Rl,
                                                float* __restrict__ z) {
  __shared__ float sB[DH * 16];        // 16 KB: R[k][col0..col0+15]
  int ct = blockIdx.x;                 // 32 col tiles
  int rg = blockIdx.y;                 // 64 row groups (256 rows each)
  int lane = threadIdx.x;
  int n = lane & 15, half = lane >> 4;
  int col0 = ct * 16;

  const float* Bg = Rl + col0;
  for (int k = lane; k < DH; k += 32)                  // 8 rows per lane
    async_row64B((uint32_t)(uintptr_t)&sB[k * 16], Bg + (size_t)k * RO);
  wait_async0();

  for (int rt = 0; rt < 16; ++rt) {
    int row0 = rg * 256 + rt * 16;
    v8f c = {};
    for (int k0 = 0; k0 < DH; k0 += 4) {
      int k = k0 + 2 * half;
      v2f a, bb;
      a[0]  = hl[(size_t)(row0 + n) * DH + k];
      a[1]  = hl[(size_t)(row0 + n) * DH + k + 1];
      bb[0] = sB[(k)     * 16 + n];
      bb[1] = sB[(k + 1) * 16 + n];
      c = wmma_f32x4(a, bb, c);
    }
    for (int r = 0; r < 8; ++r) {
      int row = row0 + r + 8 * half;
      z[(size_t)row * RO + col0 + n] = c[r];
    }
  }
}

// ------ masked softmax per node row, summed over nodes (one block/batch) ----
__global__ __launch_bounds__(512) void k_softmax_acc(const float* __restrict__ z,
                                                     float* __restrict__ acc) {
  __shared__ float red[512];
  int b = blockIdx.x;
  int t = threadIdx.x;
  float accv = 0.f;
  for (int v = 0; v < Nn; ++v) {
    float x = z[(size_t)(b * Nn + v) * RO + t];
    // mask: any(z != 0)  <=>  max(|z|) > 0
    red[t] = fabsf(x);
    __syncthreads();
    for (int s = 256; s > 0; s >>= 1) {
      if (t < s) red[t] = fmaxf(red[t], red[t + s]);
      __syncthreads();
    }
    float mxabs = red[0];
    __syncthreads();
    // softmax max
    red[t] = x;
    __syncthreads();
    for (int s = 256; s > 0; s >>= 1) {
      if (t < s) red[t] = fmaxf(red[t], red[t + s]);
      __syncthreads();
    }
    float mx = red[0];
    __syncthreads();
    float ex = expf(x - mx);
    red[t] = ex;
    __syncthreads();
    for (int s = 256; s > 0; s >>= 1) {
      if (t < s) red[t] += red[t + s];
      __syncthreads();
    }
    float sm = red[0];
    __syncthreads();
    if (mxabs > 0.f) accv += ex / sm;
  }
  acc[(size_t)b * RO + t] += accv;     // 3 sequential launches accumulate
}

// ---------------- final MLP 512->128->256->128->12 --------------------------
__global__ __launch_bounds__(128) void k_mlp(const float* __restrict__ acc,
    const float* w0, const float* b0, const float* w1, const float* b1,
    const float* w2, const float* b2, const float* w3, const float* b3,
    float* __restrict__ out) {
  __shared__ float sin_[512];
  __shared__ float s0[128];
  __shared__ float s1[256];
  __shared__ float s2[128];
  int b = blockIdx.x, t = threadIdx.x;
  for (int i = t; i < 512; i += 128) sin_[i] = acc[(size_t)b * 512 + i];
  __syncthreads();
  float v = b0[t];
  for (int i = 0; i < 512; ++i) v += sin_[i] * w0[i * 128 + t];
  s0[t] = fmaxf(v, 0.f);
  __syncthreads();
  for (int j = t; j < 256; j += 128) {
    float u = b1[j];
    for (int i = 0; i < 128; ++i) u += s0[i] * w1[i * 256 + j];
    s1[j] = fmaxf(u, 0.f);
  }
  __syncthreads();
  float w = b2[t];
  for (int i = 0; i < 256; ++i) w += s1[i] * w2[i * 128 + t];
  s2[t] = fmaxf(w, 0.f);
  __syncthreads();
  if (t < TGT) {
    float o = b3[t];
    for (int i = 0; i < 128; ++i) o += s2[i] * w3[i * TGT + t];
    out[(size_t)b * TGT + t] = o;
  }
}

// ---------------- workspace layout (bytes) ----------------------------------
#define OFF_CNT   ((size_t)0)                     // 11 ints
#define OFF_PERM  ((size_t)256)                   // 11*16384 ints = 720896 B
#define OFF_MBUF  ((size_t)721152)                // 16384*320 f   = 20971520 B
#define OFF_H1    ((size_t)21692672)              // 16384*256 f   = 16777216 B
#define OFF_H2    ((size_t)38469888)              // 16384*256 f
#define OFF_Z     ((size_t)55247104)              // 16384*512 f   = 33554432 B
#define OFF_ACC   ((size_t)88801536)              // 256*512 f

extern "C" void kernel_launch(void* const* d_in, const int* in_sizes, int n_in,
                              void* d_out, int out_size, void* d_ws, size_t ws_size,
                              hipStream_t stream) {
  const float* g    = (const float*)d_in[0];
  const float* h_in = (const float*)d_in[1];
  const float* e    = (const float*)d_in[2];
  const float* Wu0  = (const float*)d_in[3];
  const float* Wu1  = (const float*)d_in[4];
  const float* R    = (const float*)d_in[5];
  const float* fc0w = (const float*)d_in[6];
  const float* fc0b = (const float*)d_in[7];
  const float* fc1w = (const float*)d_in[8];
  const float* fc1b = (const float*)d_in[9];
  const float* fc2w = (const float*)d_in[10];
  const float* fc2b = (const float*)d_in[11];
  const float* fc3w = (const float*)d_in[12];
  const float* fc3b = (const float*)d_in[13];
  float* out = (float*)d_out;

  char* ws = (char*)d_ws;
  int*   cnt  = (int*)  (ws + OFF_CNT);
  int*   perm = (int*)  (ws + OFF_PERM);
  float* mbuf = (float*)(ws + OFF_MBUF);
  float* h1   = (float*)(ws + OFF_H1);
  float* h2   = (float*)(ws + OFF_H2);
  float* z    = (float*)(ws + OFF_Z);
  float* acc  = (float*)(ws + OFF_ACC);

  k_init<<<2048, 256, 0, stream>>>(h1 /* h1,h2 contiguous */, acc, cnt);
  k_deg<<<NODES / 256, 256, 0, stream>>>(g, cnt, perm);
  k_me<<<NODES, 64, 0, stream>>>(g, e, mbuf);

  // layer 0: m = [g@h_in | m_e];  h1 = sigmoid(m @ Wu0[deg])
  k_mh<<<dim3(16, 4, Bq), 32, 0, stream>>>(g, h_in, mbuf);
  k_update<<<dim3(16, 8, ND), 32, 0, stream>>>(mbuf, Wu0, cnt, perm, h1);
  // layer 1: m = [g@h1 | m_e];  h2 = sigmoid(m @ Wu1[deg])
  k_mh<<<dim3(16, 4, Bq), 32, 0, stream>>>(g, h1, mbuf);
  k_update<<<dim3(16, 8, ND), 32, 0, stream>>>(mbuf, Wu1, cnt, perm, h2);

  // readout over layers {h_in, h1, h2}
  const float* hs[3] = {h_in, h1, h2};
  for (int l = 0; l < 3; ++l) {
    k_readout<<<dim3(RO / 16, NODES / 256), 32, 0, stream>>>(hs[l],
                                                             R + (size_t)l * DH * RO, z);
    k_softmax_acc<<<Bq, 512, 0, stream>>>(z, acc);
  }

  k_mlp<<<Bq, 128, 0, stream>>>(acc, fc0w, fc0b, fc1w, fc1b, fc2w, fc2b,
                                fc3w, fc3b, out);
}